// NCESoftmaxLoss_sam_threshold_1477468750347
// MI455X (gfx1250) — compile-verified
//
#include <hip/hip_runtime.h>
#include <cstdint>
#include <cstddef>

#define K_DIM 4096
#define ROWS_PER_BLOCK 8
#define THREADS (ROWS_PER_BLOCK * 32)
#define CHUNK 512                    // floats per LDS chunk (2 KB)
#define NCHUNK (K_DIM / CHUNK)       // 8 chunks per row
#define F4L (CHUNK / 128)            // float4 loads per lane per chunk = 4
#define LOG2E 1.44269504088896340736f
#define LN2   0.69314718055994530942f

#define AS1 __attribute__((address_space(1)))
#define AS3 __attribute__((address_space(3)))

typedef int v4i __attribute__((vector_size(4 * sizeof(int))));

// ---- CDNA5 async global->LDS copy (ASYNCcnt path) ----------------------
__device__ __forceinline__ void async_copy_b128(const float* gsrc, float* ldst) {
#if __has_builtin(__builtin_amdgcn_global_load_async_to_lds_b128)
  __builtin_amdgcn_global_load_async_to_lds_b128(
      (AS1 v4i*)(AS1 void*)gsrc, (AS3 v4i*)(AS3 void*)ldst, 0, 0);
#else
  unsigned loff = (unsigned)(unsigned long long)(AS3 void*)ldst;
  asm volatile("global_load_async_to_lds_b128 %0, %1, off"
               :: "v"(loff), "v"(gsrc) : "memory");
#endif
}

__device__ __forceinline__ void wait_async_le4() {
#if __has_builtin(__builtin_amdgcn_s_wait_asynccnt)
  __builtin_amdgcn_s_wait_asynccnt(4);
#else
  asm volatile("s_wait_asynccnt 0x4" ::: "memory");
#endif
}

__device__ __forceinline__ void wait_async_0() {
#if __has_builtin(__builtin_amdgcn_s_wait_asynccnt)
  __builtin_amdgcn_s_wait_asynccnt(0);
#else
  asm volatile("s_wait_asynccnt 0x0" ::: "memory");
#endif
}

// ---- Kernel A: per-row online logsumexp, one HBM pass ------------------
__global__ __launch_bounds__(THREADS) void nce_loss_rows(
    const float* __restrict__ x, const float* __restrict__ thr,
    float* __restrict__ part_loss, float* __restrict__ part_cnt,
    float* __restrict__ loss0, int bsz)
{
  __shared__ float smem[ROWS_PER_BLOCK * 2 * CHUNK];   // 32 KB, double-buffered per wave
  __shared__ float wl[ROWS_PER_BLOCK];
  __shared__ float wc[ROWS_PER_BLOCK];

  const int lane = threadIdx.x & 31;        // wave32
  const int wv   = threadIdx.x >> 5;
  const int row  = blockIdx.x * ROWS_PER_BLOCK + wv;
  const float th = thr[0];

  float loss = 0.0f, msk = 0.0f;
  if (row < bsz) {
    float* b0 = smem + wv * (2 * CHUNK);
    float* b1 = b0 + CHUNK;
    const float* rp = x + (size_t)row * K_DIM;

    // prefetch chunk 0
    #pragma unroll
    for (int j = 0; j < F4L; ++j)
      async_copy_b128(rp + (j * 32 + lane) * 4, b0 + (j * 32 + lane) * 4);

    float m = -3.0e38f, s = 0.0f, x0 = 0.0f;
    for (int c = 0; c < NCHUNK; ++c) {
      float* cur = (c & 1) ? b1 : b0;
      float* nxt = (c & 1) ? b0 : b1;
      if (c + 1 < NCHUNK) {                 // issue next chunk, then wait for current
        const float* src = rp + (c + 1) * CHUNK;
        #pragma unroll
        for (int j = 0; j < F4L; ++j)
          async_copy_b128(src + (j * 32 + lane) * 4, nxt + (j * 32 + lane) * 4);
        wait_async_le4();                   // 4 newest may stay in flight
      } else {
        wait_async_0();
      }

      // load chunk into registers once (ds_load_b128), track chunk max
      float4 v[F4L];
      float cm = -3.0e38f;
      #pragma unroll
      for (int j = 0; j < F4L; ++j) {
        v[j] = *(const float4*)(cur + (j * 32 + lane) * 4);
        cm = fmaxf(cm, fmaxf(fmaxf(v[j].x, v[j].y), fmaxf(v[j].z, v[j].w)));
      }
      if (c == 0 && lane == 0) x0 = v[0].x;   // x[row][0]

      #pragma unroll
      for (int o = 16; o > 0; o >>= 1)        // cross-lane max (wave32)
        cm = fmaxf(cm, __shfl_xor(cm, o));

      const float mn = fmaxf(m, cm);
      const float mk = mn * LOG2E;
      s *= __builtin_amdgcn_exp2f((m - mn) * LOG2E);   // one rescale per chunk
      float acc = 0.0f;
      #pragma unroll
      for (int j = 0; j < F4L; ++j) {
        acc += __builtin_amdgcn_exp2f(fmaf(v[j].x, LOG2E, -mk));
        acc += __builtin_amdgcn_exp2f(fmaf(v[j].y, LOG2E, -mk));
        acc += __builtin_amdgcn_exp2f(fmaf(v[j].z, LOG2E, -mk));
        acc += __builtin_amdgcn_exp2f(fmaf(v[j].w, LOG2E, -mk));
      }
      s += acc;
      m = mn;
    }

    #pragma unroll
    for (int o = 16; o > 0; o >>= 1)          // cross-lane sum (shared max scale)
      s += __shfl_xor(s, o);

    if (lane == 0) {
      loss = m + __builtin_amdgcn_logf(s) * LN2 - x0;
      msk  = (loss > th) ? 1.0f : 0.0f;
      if (row == 0) loss0[0] = loss;
    }
  }

  if (lane == 0) { wl[wv] = loss * msk; wc[wv] = msk; }
  __syncthreads();
  if (threadIdx.x == 0) {                     // fixed-order block partial (deterministic)
    float ls = 0.0f, cs = 0.0f;
    #pragma unroll
    for (int i = 0; i < ROWS_PER_BLOCK; ++i) { ls += wl[i]; cs += wc[i]; }
    part_loss[blockIdx.x] = ls;
    part_cnt[blockIdx.x]  = cs;
  }
}

// ---- Kernel B: deterministic final reduce over block partials ----------
__global__ __launch_bounds__(32) void nce_finalize(
    const float* __restrict__ part_loss, const float* __restrict__ part_cnt,
    const float* __restrict__ loss0, float* __restrict__ out, int nblk)
{
  float ls = 0.0f, cs = 0.0f;
  for (int i = threadIdx.x; i < nblk; i += 32) { ls += part_loss[i]; cs += part_cnt[i]; }
  #pragma unroll
  for (int o = 16; o > 0; o >>= 1) { ls += __shfl_xor(ls, o); cs += __shfl_xor(cs, o); }
  if (threadIdx.x == 0)
    out[0] = (cs < 0.5f) ? loss0[0] : ls / fmaxf(cs, 1.0f);
}

extern "C" void kernel_launch(void* const* d_in, const int* in_sizes, int n_in,
                              void* d_out, int out_size, void* d_ws, size_t ws_size,
                              hipStream_t stream) {
  const float* x   = (const float*)d_in[0];
  const float* thr = (const float*)d_in[1];
  float* out = (float*)d_out;

  const int bsz  = in_sizes[0] / K_DIM;                       // 32768
  const int nblk = (bsz + ROWS_PER_BLOCK - 1) / ROWS_PER_BLOCK; // 4096

  float* part_loss = (float*)d_ws;          // every slot written each call
  float* part_cnt  = part_loss + nblk;
  float* loss0     = part_cnt + nblk;

  nce_loss_rows<<<nblk, THREADS, 0, stream>>>(x, thr, part_loss, part_cnt, loss0, bsz);
  nce_finalize<<<1, 32, 0, stream>>>(part_loss, part_cnt, loss0, out, nblk);
}